// EnhancedEncoder_3006477107885
// MI455X (gfx1250) — compile-verified
//
#include <hip/hip_runtime.h>

#define EPS 1e-5f
#define B_GROUPS 1024
#define NPTS 32
#define NCOLS 32768          // B_GROUPS * NPTS
#define C1 128
#define C2 256
#define C3 512
#define C4 384
#define KNN 8

typedef __attribute__((ext_vector_type(2))) float v2f;
typedef __attribute__((ext_vector_type(8))) float v8f;

__device__ __forceinline__ v8f wmma_f32(v2f a, v2f b, v8f c) {
    return __builtin_amdgcn_wmma_f32_16x16x4_f32(
        false, a, false, b, (short)0, c, false, false);
}

// ---------------------------------------------------------------------------
// conv1: f1[o][n] = b1[o] + W1[o]*pg[n] (K=3), n fastest for coalescing
// ---------------------------------------------------------------------------
__global__ void __launch_bounds__(256)
k_conv1(const float* __restrict__ pg, const float* __restrict__ W1,
        const float* __restrict__ b1, float* __restrict__ f1)
{
    int tid = blockIdx.x * 256 + threadIdx.x;     // NCOLS*C1 threads
    int n = tid & (NCOLS - 1);
    int o = tid >> 15;
    float x = pg[n * 3 + 0], y = pg[n * 3 + 1], z = pg[n * 3 + 2];
    f1[(size_t)o * NCOLS + n] =
        b1[o] + W1[o * 3 + 0] * x + W1[o * 3 + 1] * y + W1[o * 3 + 2] * z;
}

// ---------------------------------------------------------------------------
// BN stats for one channel -> bn(x) = x*s[c] + t[c].  Deterministic.
// ---------------------------------------------------------------------------
__global__ void __launch_bounds__(256)
k_bnstats(const float* __restrict__ X, int Ncols,
          const float* __restrict__ gamma, const float* __restrict__ beta,
          float* __restrict__ s_out, float* __restrict__ t_out)
{
    __shared__ float r1[256], r2[256];
    int c = blockIdx.x;
    const float* col = X + (size_t)c * Ncols;
    float a = 0.f, b = 0.f;
    for (int i = threadIdx.x; i < Ncols; i += 256) {
        float v = col[i];
        a += v; b += v * v;
    }
    r1[threadIdx.x] = a; r2[threadIdx.x] = b;
    __syncthreads();
    for (int off = 128; off > 0; off >>= 1) {
        if (threadIdx.x < off) {
            r1[threadIdx.x] += r1[threadIdx.x + off];
            r2[threadIdx.x] += r2[threadIdx.x + off];
        }
        __syncthreads();
    }
    if (threadIdx.x == 0) {
        float inv = 1.f / (float)Ncols;
        float mu  = r1[0] * inv;
        float var = r2[0] * inv - mu * mu;
        float sc  = rsqrtf(var + EPS) * gamma[c];
        s_out[c] = sc;
        t_out[c] = beta[c] - mu * sc;
    }
}

// ---------------------------------------------------------------------------
// max over the 32 points of each group:  out[b*C+c] = max_l X[c][b*32+l]
// ---------------------------------------------------------------------------
__global__ void __launch_bounds__(256)
k_maxpool(const float* __restrict__ X, float* __restrict__ out, int C)
{
    int tid = blockIdx.x * 256 + threadIdx.x;
    int c = tid % C;
    int b = tid / C;
    const float* col = X + (size_t)c * NCOLS + b * NPTS;
    float m = col[0];
#pragma unroll
    for (int l = 1; l < NPTS; ++l) m = fmaxf(m, col[l]);
    out[b * C + c] = m;
}

// ---------------------------------------------------------------------------
// Specialized f32 WMMA GEMM:  OUT[o][n] = sum_k W[o][k]*in(k,n) + bias[o]
//   MODE 1: in = relu(IN*s+t)            (single segment, SPLIT==K)
//   MODE 2: in = [fg broadcast | f2]     (segment split at SPLIT=C2)
//   MODE 3: in = [fg2 | a*mdx*inv + b]   (segment split at SPLIT=C4)
// 8 waves/block; each wave owns a 16(M) x 64(N) strip = 4 D tiles, one
// A-fetch (b64) feeds 4 v_wmma_f32_16x16x4_f32.  Block tile: 128 x 64.
// ---------------------------------------------------------------------------
template<int MODE, int K, int SPLIT, int NC>
__global__ void __launch_bounds__(256)
k_gemm(const float* __restrict__ W, const float* __restrict__ bias,
       const float* __restrict__ IN, float* __restrict__ OUT,
       const float* __restrict__ s, const float* __restrict__ t,
       const float* __restrict__ fgbuf, const float* __restrict__ aux,
       const float* __restrict__ alpha, const float* __restrict__ betaA,
       const float* __restrict__ stdp)
{
    __shared__ float sh_s[K];
    __shared__ float sh_t[K];
    if constexpr (MODE == 1) {
        for (int i = threadIdx.x; i < K; i += 256) { sh_s[i] = s[i]; sh_t[i] = t[i]; }
        __syncthreads();
    } else if constexpr (MODE == 3) {
        float inv = 1.f / (stdp[0] + EPS);
        for (int i = threadIdx.x; i < C4; i += 256) {
            sh_s[i] = alpha[i] * inv;
            sh_t[i] = betaA[i];
        }
        __syncthreads();
    }

    const int lane  = threadIdx.x & 31;
    const int wave  = threadIdx.x >> 5;
    const int n0    = blockIdx.x << 6;                 // 64 columns / block
    const int o0    = (blockIdx.y << 7) + (wave << 4); // 128 rows / block
    const int mrow  = lane & 15;
    const int khalf = (lane >> 4) << 1;                // 0 or 2
    const int c0    = n0 + mrow;
    const float* wrow = W + (size_t)(o0 + mrow) * K + khalf;

    v8f acc0 = {}, acc1 = {}, acc2 = {}, acc3 = {};

    // ---- segment 1: k in [0, SPLIT) ----
#pragma unroll 4
    for (int kb = 0; kb < SPLIT; kb += 4) {
        __builtin_prefetch(wrow + kb + 64, 0, 3);
        v2f a; a.x = wrow[kb]; a.y = wrow[kb + 1];
        const int r0 = kb + khalf, r1 = r0 + 1;
        v2f b0, b1, b2, b3;
        if constexpr (MODE == 1) {
            const float* p0 = IN + (size_t)r0 * NC + c0;
            const float* p1 = IN + (size_t)r1 * NC + c0;
            float s0 = sh_s[r0], t0 = sh_t[r0];
            float s1 = sh_s[r1], t1 = sh_t[r1];
            b0.x = fmaxf(fmaf(p0[ 0], s0, t0), 0.f); b0.y = fmaxf(fmaf(p1[ 0], s1, t1), 0.f);
            b1.x = fmaxf(fmaf(p0[16], s0, t0), 0.f); b1.y = fmaxf(fmaf(p1[16], s1, t1), 0.f);
            b2.x = fmaxf(fmaf(p0[32], s0, t0), 0.f); b2.y = fmaxf(fmaf(p1[32], s1, t1), 0.f);
            b3.x = fmaxf(fmaf(p0[48], s0, t0), 0.f); b3.y = fmaxf(fmaf(p1[48], s1, t1), 0.f);
        } else if constexpr (MODE == 2) {
            const float* g0 = fgbuf + ((c0      ) >> 5) * C2;
            const float* g1 = fgbuf + ((c0 + 16 ) >> 5) * C2;
            const float* g2 = fgbuf + ((c0 + 32 ) >> 5) * C2;
            const float* g3 = fgbuf + ((c0 + 48 ) >> 5) * C2;
            b0.x = g0[r0]; b0.y = g0[r1];
            b1.x = g1[r0]; b1.y = g1[r1];
            b2.x = g2[r0]; b2.y = g2[r1];
            b3.x = g3[r0]; b3.y = g3[r1];
        } else { // MODE == 3
            const float* g0 = fgbuf + (size_t)(c0      ) * C4;
            const float* g1 = fgbuf + (size_t)(c0 + 16 ) * C4;
            const float* g2 = fgbuf + (size_t)(c0 + 32 ) * C4;
            const float* g3 = fgbuf + (size_t)(c0 + 48 ) * C4;
            b0.x = g0[r0]; b0.y = g0[r1];
            b1.x = g1[r0]; b1.y = g1[r1];
            b2.x = g2[r0]; b2.y = g2[r1];
            b3.x = g3[r0]; b3.y = g3[r1];
        }
        acc0 = wmma_f32(a, b0, acc0);
        acc1 = wmma_f32(a, b1, acc1);
        acc2 = wmma_f32(a, b2, acc2);
        acc3 = wmma_f32(a, b3, acc3);
    }

    // ---- segment 2: k in [SPLIT, K) ----
    if constexpr (SPLIT < K) {
#pragma unroll 4
        for (int kb = SPLIT; kb < K; kb += 4) {
            __builtin_prefetch(wrow + kb + 64, 0, 3);
            v2f a; a.x = wrow[kb]; a.y = wrow[kb + 1];
            const int r0 = kb + khalf, r1 = r0 + 1;
            v2f b0, b1, b2, b3;
            if constexpr (MODE == 2) {
                const float* p0 = aux + (size_t)(r0 - C2) * NC + c0;
                const float* p1 = aux + (size_t)(r1 - C2) * NC + c0;
                b0.x = p0[ 0]; b0.y = p1[ 0];
                b1.x = p0[16]; b1.y = p1[16];
                b2.x = p0[32]; b2.y = p1[32];
                b3.x = p0[48]; b3.y = p1[48];
            } else { // MODE == 3
                const int cc0 = r0 - C4, cc1 = r1 - C4;
                float sa0 = sh_s[cc0], ta0 = sh_t[cc0];
                float sa1 = sh_s[cc1], ta1 = sh_t[cc1];
                const float* m0 = aux + (size_t)(c0      ) * C4;
                const float* m1 = aux + (size_t)(c0 + 16 ) * C4;
                const float* m2 = aux + (size_t)(c0 + 32 ) * C4;
                const float* m3 = aux + (size_t)(c0 + 48 ) * C4;
                b0.x = fmaf(m0[cc0], sa0, ta0); b0.y = fmaf(m0[cc1], sa1, ta1);
                b1.x = fmaf(m1[cc0], sa0, ta0); b1.y = fmaf(m1[cc1], sa1, ta1);
                b2.x = fmaf(m2[cc0], sa0, ta0); b2.y = fmaf(m2[cc1], sa1, ta1);
                b3.x = fmaf(m3[cc0], sa0, ta0); b3.y = fmaf(m3[cc1], sa1, ta1);
            }
            acc0 = wmma_f32(a, b0, acc0);
            acc1 = wmma_f32(a, b1, acc1);
            acc2 = wmma_f32(a, b2, acc2);
            acc3 = wmma_f32(a, b3, acc3);
        }
    }

    // ---- store D tiles (+bias): lane L holds N=L%16, VGPR v -> M=v+8*(L>=16)
    const int mbase = o0 + ((lane >> 4) << 3);
#pragma unroll
    for (int v = 0; v < 8; ++v) {
        int m = mbase + v;
        float bv = bias[m];
        size_t row = (size_t)m * NC;
        OUT[row + c0     ] = acc0[v] + bv;
        OUT[row + c0 + 16] = acc1[v] + bv;
        OUT[row + c0 + 32] = acc2[v] + bv;
        OUT[row + c0 + 48] = acc3[v] + bv;
    }
}

// ---------------------------------------------------------------------------
// KNN: one wave per group; d = s_n + s_m - 2*dot (reference formula; strict <
// keeps the lowest index on ties, matching top_k stability).
// ---------------------------------------------------------------------------
__global__ void __launch_bounds__(32)
k_knn(const float* __restrict__ pg, int* __restrict__ idx)
{
    __shared__ float px[32], py[32], pz[32], sq[32];
    int b = blockIdx.x;
    int n = threadIdx.x;
    float x = pg[(b * 32 + n) * 3 + 0];
    float y = pg[(b * 32 + n) * 3 + 1];
    float z = pg[(b * 32 + n) * 3 + 2];
    px[n] = x; py[n] = y; pz[n] = z;
    sq[n] = x * x + y * y + z * z;
    __syncthreads();
    float sn = sq[n];
    float d[32];
#pragma unroll
    for (int m = 0; m < 32; ++m)
        d[m] = sn + sq[m] - 2.f * (x * px[m] + y * py[m] + z * pz[m]);
#pragma unroll
    for (int k = 0; k < KNN; ++k) {
        int bi = 0; float bv = d[0];
#pragma unroll
        for (int m = 1; m < 32; ++m)
            if (d[m] < bv) { bv = d[m]; bi = m; }
        idx[(b * 32 + n) * KNN + k] = bi;
        d[bi] = 3.4e38f;
    }
}

// ---------------------------------------------------------------------------
// meanDx[b][c] = mean_{n,k}(feat[idx]-feat) + per-block (sum,sumsq) partials
// ---------------------------------------------------------------------------
__global__ void __launch_bounds__(256)
k_meandx(const float* __restrict__ f4, const int* __restrict__ idx,
         float* __restrict__ meanDx, float* __restrict__ stdpart)
{
    __shared__ int   sidx[256];
    __shared__ float r1[256], r2[256];
    int b = blockIdx.x;
    sidx[threadIdx.x] = idx[b * 256 + threadIdx.x];
    __syncthreads();
    float ls1 = 0.f, ls2 = 0.f;
    for (int c = threadIdx.x; c < C4; c += 256) {
        const float* col = f4 + (size_t)c * NCOLS + b * NPTS;
        float acc = 0.f;
#pragma unroll 4
        for (int n = 0; n < NPTS; ++n) {
            float base = col[n];
#pragma unroll
            for (int k = 0; k < KNN; ++k) {
                float dx = col[sidx[n * KNN + k]] - base;
                acc += dx;
                ls2 += dx * dx;
            }
        }
        ls1 += acc;
        meanDx[b * C4 + c] = acc * (1.f / (NPTS * KNN));
    }
    r1[threadIdx.x] = ls1; r2[threadIdx.x] = ls2;
    __syncthreads();
    for (int off = 128; off > 0; off >>= 1) {
        if (threadIdx.x < off) {
            r1[threadIdx.x] += r1[threadIdx.x + off];
            r2[threadIdx.x] += r2[threadIdx.x + off];
        }
        __syncthreads();
    }
    if (threadIdx.x == 0) { stdpart[2 * b] = r1[0]; stdpart[2 * b + 1] = r2[0]; }
}

__global__ void __launch_bounds__(256)
k_stdfinal(const float* __restrict__ stdpart, float* __restrict__ stdval)
{
    __shared__ float r1[256], r2[256];
    float a = 0.f, b = 0.f;
    for (int i = threadIdx.x; i < B_GROUPS; i += 256) {
        a += stdpart[2 * i]; b += stdpart[2 * i + 1];
    }
    r1[threadIdx.x] = a; r2[threadIdx.x] = b;
    __syncthreads();
    for (int off = 128; off > 0; off >>= 1) {
        if (threadIdx.x < off) {
            r1[threadIdx.x] += r1[threadIdx.x + off];
            r2[threadIdx.x] += r2[threadIdx.x + off];
        }
        __syncthreads();
    }
    if (threadIdx.x == 0) {
        double cnt = (double)B_GROUPS * NPTS * KNN * C4;   // 100,663,296
        double S1 = r1[0], S2 = r2[0];
        double var = (S2 - S1 * S1 / cnt) / (cnt - 1.0);   // ddof=1
        stdval[0] = (float)sqrt(var > 0.0 ? var : 0.0);
    }
}

// ---------------------------------------------------------------------------
__global__ void __launch_bounds__(256)
k_final(const float* __restrict__ ff, const float* __restrict__ sf,
        const float* __restrict__ tf, float* __restrict__ out)
{
    int tid = blockIdx.x * 256 + threadIdx.x;     // B_GROUPS*C4
    int c = tid % C4;
    int b = tid / C4;
    out[tid] = fmaxf(fmaf(ff[(size_t)c * B_GROUPS + b], sf[c], tf[c]), 0.f);
}

// ---------------------------------------------------------------------------
extern "C" void kernel_launch(void* const* d_in, const int* in_sizes, int n_in,
                              void* d_out, int out_size, void* d_ws, size_t ws_size,
                              hipStream_t stream)
{
    const float* pg    = (const float*)d_in[0];
    const float* W1    = (const float*)d_in[1];
    const float* b1    = (const float*)d_in[2];
    const float* g1    = (const float*)d_in[3];
    const float* be1   = (const float*)d_in[4];
    const float* W2    = (const float*)d_in[5];
    const float* b2    = (const float*)d_in[6];
    const float* W3    = (const float*)d_in[7];
    const float* b3    = (const float*)d_in[8];
    const float* g3    = (const float*)d_in[9];
    const float* be3   = (const float*)d_in[10];
    const float* W4    = (const float*)d_in[11];
    const float* b4    = (const float*)d_in[12];
    const float* alpha = (const float*)d_in[13];
    const float* betaA = (const float*)d_in[14];
    const float* Wf    = (const float*)d_in[15];
    const float* bf    = (const float*)d_in[16];
    const float* gamf  = (const float*)d_in[17];
    const float* betf  = (const float*)d_in[18];

    float* ws = (float*)d_ws;
    size_t o = 0;
    float* f1  = ws + o; o += (size_t)C1 * NCOLS;
    float* f2  = ws + o; o += (size_t)C2 * NCOLS;
    float* f3  = ws + o; o += (size_t)C3 * NCOLS;
    float* f4  = ws + o; o += (size_t)C4 * NCOLS;
    float* fg  = ws + o; o += (size_t)B_GROUPS * C2;
    float* fg2 = ws + o; o += (size_t)B_GROUPS * C4;
    float* mdx = ws + o; o += (size_t)B_GROUPS * C4;
    float* ff  = ws + o; o += (size_t)C4 * B_GROUPS;
    float* s1  = ws + o; o += C1;   float* t1 = ws + o; o += C1;
    float* s3  = ws + o; o += C3;   float* t3 = ws + o; o += C3;
    float* sf  = ws + o; o += C4;   float* tf = ws + o; o += C4;
    float* stdpart = ws + o; o += 2 * B_GROUPS;
    float* stdval  = ws + o; o += 4;
    int*   idx = (int*)(ws + o);

    // Stage 1: conv1 -> BN1 stats
    k_conv1<<<(C1 * NCOLS) / 256, 256, 0, stream>>>(pg, W1, b1, f1);
    k_bnstats<<<C1, 256, 0, stream>>>(f1, NCOLS, g1, be1, s1, t1);

    // Stage 2: conv2 (K=128) with fused BN1+ReLU
    k_gemm<1, C1, C1, NCOLS><<<dim3(NCOLS / 64, C2 / 128), 256, 0, stream>>>(
        W2, b2, f1, f2, s1, t1, nullptr, nullptr, nullptr, nullptr, nullptr);
    k_maxpool<<<(B_GROUPS * C2) / 256, 256, 0, stream>>>(f2, fg, C2);

    // Stage 3: conv3 (K=512) over virtual concat [fg | f2]
    k_gemm<2, C3, C2, NCOLS><<<dim3(NCOLS / 64, C3 / 128), 256, 0, stream>>>(
        W3, b3, nullptr, f3, nullptr, nullptr, fg, f2, nullptr, nullptr, nullptr);
    k_bnstats<<<C3, 256, 0, stream>>>(f3, NCOLS, g3, be3, s3, t3);

    // Stage 4: conv4 (K=512) with fused BN3+ReLU
    k_gemm<1, C3, C3, NCOLS><<<dim3(NCOLS / 64, C4 / 128), 256, 0, stream>>>(
        W4, b4, f3, f4, s3, t3, nullptr, nullptr, nullptr, nullptr, nullptr);
    k_maxpool<<<(B_GROUPS * C4) / 256, 256, 0, stream>>>(f4, fg2, C4);

    // Stage 5: geometric-affine branch
    k_knn<<<B_GROUPS, 32, 0, stream>>>(pg, idx);
    k_meandx<<<B_GROUPS, 256, 0, stream>>>(f4, idx, mdx, stdpart);
    k_stdfinal<<<1, 256, 0, stream>>>(stdpart, stdval);

    // Stage 6: convf (K=768, N=1024) over [fg2 | affine(meanDx)]
    k_gemm<3, 2 * C4, C4, B_GROUPS><<<dim3(B_GROUPS / 64, C4 / 128), 256, 0, stream>>>(
        Wf, bf, nullptr, ff, nullptr, nullptr, fg2, mdx, alpha, betaA, stdval);
    k_bnstats<<<C4, 256, 0, stream>>>(ff, B_GROUPS, gamf, betf, sf, tf);
    k_final<<<(B_GROUPS * C4) / 256, 256, 0, stream>>>(ff, sf, tf, (float*)d_out);
}